// PanopticDeepLab_70342974374300
// MI455X (gfx1250) — compile-verified
//
#include <hip/hip_runtime.h>
#include <stdint.h>
#include <math.h>

// Problem constants (fixed by the reference harness)
namespace {
constexpr int kB  = 2;
constexpr int kC  = 19;
constexpr int kH  = 512;
constexpr int kW  = 1024;
constexpr int kHW = kH * kW;
constexpr int kTOP = 200;
constexpr int kCAP = 16384;          // peak-list capacity per batch (~9.6K expected)
constexpr float kTHR = 0.1f;
constexpr int kMINAREA = 32;
constexpr int kIDBASE = 1000;

constexpr int TX = 64, TY = 8, HAL = 3;
constexpr int TWD = TX + 2 * HAL;    // 70
constexpr int THT = TY + 2 * HAL;    // 14
}

// ---------------------------------------------------------------------------
// K1: per-pixel argmax over 19 channels -> sem_pred written into d_out (int32)
// float4-vectorized: each thread owns 4 consecutive pixels. Strict '>' keeps
// the FIRST max index (jnp.argmax semantics).
// ---------------------------------------------------------------------------
__global__ void argmax_kernel(const float* __restrict__ logits,
                              int* __restrict__ out) {
  int t = blockIdx.x * blockDim.x + threadIdx.x;    // quad index
  const int total = kB * kHW / 4;
  if (t >= total) return;
  int b = t / (kHW / 4);
  int rem4 = t - b * (kHW / 4);
  const float4* base = (const float4*)logits + (size_t)b * kC * (kHW / 4) + rem4;
  float4 best = base[0];
  int4 bi = make_int4(0, 0, 0, 0);
#pragma unroll
  for (int c = 1; c < kC; ++c) {
    float4 v = base[(size_t)c * (kHW / 4)];
    if (v.x > best.x) { best.x = v.x; bi.x = c; }
    if (v.y > best.y) { best.y = v.y; bi.y = c; }
    if (v.z > best.z) { best.z = v.z; bi.z = c; }
    if (v.w > best.w) { best.w = v.w; bi.w = c; }
  }
  ((int4*)out)[t] = bi;
}

// ---------------------------------------------------------------------------
// K2: 7x7 max-pool NMS peak detection. Heat tile (+halo 3, clamp-pad) is
// staged into LDS with CDNA5 async global->LDS copies, then each thread tests
// its pixel. Peaks appended as sortable u64 keys:
//   key = (float_bits(score) << 32) | (0xFFFFFFFF - linear_idx)
// (descending key order == jax.lax.top_k order: score desc, index asc)
// ---------------------------------------------------------------------------
__global__ void peaks_kernel(const float* __restrict__ heat,
                             unsigned long long* __restrict__ keys,
                             int* __restrict__ cnt) {
  __shared__ float tile[TWD * THT];
  const int b = blockIdx.z;
  const int bx = blockIdx.x * TX, by = blockIdx.y * TY;
  const int tid = threadIdx.y * TX + threadIdx.x;
  const float* hb = heat + (size_t)b * kHW;
  // low 32 bits of a generic pointer to LDS == wave-relative LDS byte address
  unsigned ldsbase = (unsigned)(uintptr_t)(void*)tile;

  for (int i = tid; i < TWD * THT; i += TX * TY) {
    int ty = i / TWD, tx = i - ty * TWD;
    int gy = by + ty - HAL; gy = gy < 0 ? 0 : (gy > kH - 1 ? kH - 1 : gy);
    int gx = bx + tx - HAL; gx = gx < 0 ? 0 : (gx > kW - 1 ? kW - 1 : gx);
    unsigned long long gaddr =
        (unsigned long long)(uintptr_t)(hb + (size_t)gy * kW + gx);
    unsigned laddr = ldsbase + 4u * (unsigned)i;
    asm volatile("global_load_async_to_lds_b32 %0, %1, off"
                 :: "v"(laddr), "v"(gaddr) : "memory");
  }
  asm volatile("s_wait_asynccnt 0" ::: "memory");
  __syncthreads();

  const int lx = threadIdx.x + HAL, ly = threadIdx.y + HAL;
  float c = tile[ly * TWD + lx];
  float m = -1e30f;
#pragma unroll
  for (int dy = -HAL; dy <= HAL; ++dy)
#pragma unroll
    for (int dx = -HAL; dx <= HAL; ++dx) {
      float v = tile[(ly + dy) * TWD + (lx + dx)];
      m = v > m ? v : m;
    }
  if (c > kTHR && c == m) {
    unsigned idx = (unsigned)((by + threadIdx.y) * kW + (bx + threadIdx.x));
    unsigned long long key =
        ((unsigned long long)__float_as_uint(c) << 32) | (0xFFFFFFFFu - idx);
    int slot = atomicAdd(&cnt[b], 1);
    if (slot < kCAP) keys[(size_t)b * kCAP + slot] = key;
  }
}

// ---------------------------------------------------------------------------
// K3: exact top-200 selection per batch (one 1024-thread block). 16 keys per
// thread live in registers; 200 rounds of block-max via wave shuffles. Keys
// are globally unique, so the winner is unique; clear-by-value avoids any
// dynamic register indexing (no scratch spills). Then slots 0..199 set up the
// center metadata and the (cy,cx) -> k map.
// ---------------------------------------------------------------------------
__global__ void __launch_bounds__(1024)
topk_kernel(const unsigned long long* __restrict__ keys,
            const int* __restrict__ cnt,
            const int* __restrict__ sem,          // d_out (sem_pred)
            short* __restrict__ map,
            int* __restrict__ cyv, int* __restrict__ cxv,
            int* __restrict__ thing, int* __restrict__ vflag) {
  const int b = blockIdx.x;
  const int tid = threadIdx.x;
  int n = cnt[b]; if (n > kCAP) n = kCAP;
  const unsigned long long* kb = keys + (size_t)b * kCAP;

  unsigned long long lk[16];
#pragma unroll
  for (int j = 0; j < 16; ++j) {
    int i = tid + j * 1024;
    lk[j] = (i < n) ? kb[i] : 0ull;
  }

  __shared__ unsigned long long warpmax[32];
  __shared__ unsigned long long blockmax;
  __shared__ unsigned long long sel[kTOP];
  const int lane = tid & 31, wid = tid >> 5;

  for (int k = 0; k < kTOP; ++k) {
    unsigned long long lm = 0ull;
#pragma unroll
    for (int j = 0; j < 16; ++j) lm = lk[j] > lm ? lk[j] : lm;
    unsigned long long wm = lm;
#pragma unroll
    for (int off = 16; off >= 1; off >>= 1) {
      unsigned long long o = __shfl_xor(wm, off, 32);
      wm = o > wm ? o : wm;
    }
    if (lane == 0) warpmax[wid] = wm;
    __syncthreads();
    if (wid == 0) {
      unsigned long long v = warpmax[lane];
#pragma unroll
      for (int off = 16; off >= 1; off >>= 1) {
        unsigned long long o = __shfl_xor(v, off, 32);
        v = o > v ? o : v;
      }
      if (lane == 0) blockmax = v;
    }
    __syncthreads();
    unsigned long long bm = blockmax;
    if (bm != 0ull) {
      if (lm == bm) sel[k] = bm;                 // unique winner records it
#pragma unroll
      for (int j = 0; j < 16; ++j) if (lk[j] == bm) lk[j] = 0ull;
    } else if (tid == 0) {
      sel[k] = 0ull;                              // ran out of peaks
    }
    __syncthreads();
  }

  if (tid < kTOP) {
    unsigned long long key = sel[tid];
    const int o = b * kTOP + tid;
    if (key != 0ull) {
      unsigned idx = 0xFFFFFFFFu - (unsigned)(key & 0xFFFFFFFFull);
      int cls = sem[(size_t)b * kHW + idx];
      cyv[o] = (int)(idx / kW);
      cxv[o] = (int)(idx % kW);
      thing[o] = (cls >= 11 && cls <= 18) ? 1 : 0;   // THING_CLASSES = 11..18
      vflag[o] = 1;
      map[(size_t)b * kHW + idx] = (short)tid;
    } else {
      cyv[o] = -1; cxv[o] = -1; thing[o] = 0; vflag[o] = 0;
    }
  }
}

// ---------------------------------------------------------------------------
// K4: mask areas. A pixel's shifted coord (sy,sx) can cover only integer
// centers strictly inside its unit disk: iy in {floor(sy), floor(sy)+1},
// same for ix -> <=4 map lookups per pixel. __f*_rn forces mul-mul-add
// rounding identical to the reference (no FMA contraction on the d2<1 test).
// ---------------------------------------------------------------------------
__global__ void area_kernel(const float* __restrict__ off,
                            const short* __restrict__ map,
                            int* __restrict__ area) {
  int t = blockIdx.x * blockDim.x + threadIdx.x;
  if (t >= kB * kHW) return;
  int b = t / kHW, p = t - b * kHW;
  int y = p / kW, x = p - y * kW;
  const float* ob = off + (size_t)b * 2 * kHW;
  float sy = (float)y + ob[p];
  float sx = (float)x + ob[kHW + p];
  int fy = (int)floorf(sy), fx = (int)floorf(sx);
  const short* mb = map + (size_t)b * kHW;
#pragma unroll
  for (int iy = fy; iy <= fy + 1; ++iy) {
    if (iy < 0 || iy >= kH) continue;
    float dy = __fadd_rn(sy, -(float)iy);
#pragma unroll
    for (int ix = fx; ix <= fx + 1; ++ix) {
      if (ix < 0 || ix >= kW) continue;
      short k = mb[iy * kW + ix];
      if (k < 0) continue;
      float dx = __fadd_rn(sx, -(float)ix);
      float d2 = __fadd_rn(__fmul_rn(dy, dy), __fmul_rn(dx, dx));
      if (d2 < 1.0f) atomicAdd(&area[b * kTOP + k], 1);
    }
  }
}

// ---------------------------------------------------------------------------
// K5: vk = valid & thing & (area>=32); exclusive prefix sum -> instance ids.
// ---------------------------------------------------------------------------
__global__ void flags_kernel(const int* __restrict__ vflag,
                             const int* __restrict__ thing,
                             const int* __restrict__ area,
                             int* __restrict__ vk,
                             int* __restrict__ finalid) {
  const int b = blockIdx.x, k = threadIdx.x;   // blockDim = 256
  __shared__ int s[256];
  int v = 0;
  if (k < kTOP) {
    int o = b * kTOP + k;
    v = (vflag[o] && thing[o] && area[o] >= kMINAREA) ? 1 : 0;
  }
  s[k] = v;
  __syncthreads();
  for (int off = 1; off < 256; off <<= 1) {
    int add = (k >= off) ? s[k - off] : 0;
    __syncthreads();
    s[k] += add;
    __syncthreads();
  }
  if (k < kTOP) {
    int o = b * kTOP + k;
    vk[o] = v;
    finalid[o] = kIDBASE + (s[k] - v);          // exclusive count of prior vk
  }
}

// ---------------------------------------------------------------------------
// K6: final fusion. Highest slot k among covering valid masks wins
// ("sequential overwrite": last valid mask = max k). Uncovered keeps sem_pred.
// ---------------------------------------------------------------------------
__global__ void assign_kernel(const float* __restrict__ off,
                              const short* __restrict__ map,
                              const int* __restrict__ vk,
                              const int* __restrict__ finalid,
                              int* __restrict__ out) {
  int t = blockIdx.x * blockDim.x + threadIdx.x;
  if (t >= kB * kHW) return;
  int b = t / kHW, p = t - b * kHW;
  int y = p / kW, x = p - y * kW;
  const float* ob = off + (size_t)b * 2 * kHW;
  float sy = (float)y + ob[p];
  float sx = (float)x + ob[kHW + p];
  int fy = (int)floorf(sy), fx = (int)floorf(sx);
  const short* mb = map + (size_t)b * kHW;
  int best = -1;
#pragma unroll
  for (int iy = fy; iy <= fy + 1; ++iy) {
    if (iy < 0 || iy >= kH) continue;
    float dy = __fadd_rn(sy, -(float)iy);
#pragma unroll
    for (int ix = fx; ix <= fx + 1; ++ix) {
      if (ix < 0 || ix >= kW) continue;
      short k = mb[iy * kW + ix];
      if (k < 0 || !vk[b * kTOP + k]) continue;
      float dx = __fadd_rn(sx, -(float)ix);
      float d2 = __fadd_rn(__fmul_rn(dy, dy), __fmul_rn(dx, dx));
      if (d2 < 1.0f && (int)k > best) best = (int)k;
    }
  }
  if (best >= 0) out[t] = finalid[b * kTOP + best];
}

// ---------------------------------------------------------------------------
extern "C" void kernel_launch(void* const* d_in, const int* in_sizes, int n_in,
                              void* d_out, int out_size, void* d_ws,
                              size_t ws_size, hipStream_t stream) {
  (void)in_sizes; (void)n_in; (void)out_size; (void)ws_size;
  const float* logits = (const float*)d_in[0];   // [B,19,H,W]
  const float* heat   = (const float*)d_in[1];   // [B,1,H,W]
  const float* off    = (const float*)d_in[2];   // [B,2,H,W]
  int* out = (int*)d_out;                        // [B,H,W] int32

  // workspace carve-out (256B aligned regions), ~2.3 MB total
  char* ws = (char*)d_ws;
  size_t o = 0;
  auto carve = [&](size_t bytes) {
    char* p = ws + o;
    o = (o + bytes + 255) & ~(size_t)255;
    return p;
  };
  int*                 cnt     = (int*)carve((size_t)kB * 4);
  unsigned long long*  keys    = (unsigned long long*)carve((size_t)kB * kCAP * 8);
  short*               map     = (short*)carve((size_t)kB * kHW * 2);
  int*                 cy      = (int*)carve((size_t)kB * kTOP * 4);
  int*                 cx      = (int*)carve((size_t)kB * kTOP * 4);
  int*                 thing   = (int*)carve((size_t)kB * kTOP * 4);
  int*                 vflag   = (int*)carve((size_t)kB * kTOP * 4);
  int*                 area    = (int*)carve((size_t)kB * kTOP * 4);
  int*                 vk      = (int*)carve((size_t)kB * kTOP * 4);
  int*                 finalid = (int*)carve((size_t)kB * kTOP * 4);

  // re-initialize scratch every call (graph-replay safe, deterministic)
  hipMemsetAsync(cnt, 0, (size_t)kB * 4, stream);
  hipMemsetAsync(map, 0xFF, (size_t)kB * kHW * 2, stream);   // -1 everywhere
  hipMemsetAsync(area, 0, (size_t)kB * kTOP * 4, stream);

  argmax_kernel<<<(kB * kHW / 4) / 256, 256, 0, stream>>>(logits, out);

  dim3 pblk(TX, TY, 1), pgrd(kW / TX, kH / TY, kB);
  peaks_kernel<<<pgrd, pblk, 0, stream>>>(heat, keys, cnt);

  topk_kernel<<<kB, 1024, 0, stream>>>(keys, cnt, out, map, cy, cx, thing, vflag);

  area_kernel<<<(kB * kHW + 255) / 256, 256, 0, stream>>>(off, map, area);

  flags_kernel<<<kB, 256, 0, stream>>>(vflag, thing, area, vk, finalid);

  assign_kernel<<<(kB * kHW + 255) / 256, 256, 0, stream>>>(off, map, vk,
                                                            finalid, out);
}